// BktModel_54571854463440
// MI455X (gfx1250) — compile-verified
//
#include <hip/hip_runtime.h>
#include <hip/hip_bf16.h>

typedef float v2f __attribute__((ext_vector_type(2)));
typedef float v8f __attribute__((ext_vector_type(8)));

#define B_N 256
#define T_N 1000
#define C_N 100
#define D_N 256

// numerically-stable 2-way logsumexp
__device__ __forceinline__ float lse2(float x, float y) {
    float m = fmaxf(x, y);
    float d = fminf(x, y) - m;          // <= 0
    return m + log1pf(__expf(d));
}

// full-wave32 butterfly sum: all lanes end with the total
__device__ __forceinline__ float wred32(float v) {
#pragma unroll
    for (int off = 16; off > 0; off >>= 1)
        v += __shfl_xor(v, off, 32);
    return v;
}

// ---------------------------------------------------------------------------
// Kernel 1: parameter tables (tiny).
//   H = tanh(kc_reps @ W1 + b1) via V_WMMA_F32_16X16X4_F32 (exact f32);
//   kl = H @ W2 + b2; assemble + log-softmax into tbl (d_ws):
//     tbl[      c*4 + s*2 + o] = log_obs[c][s][o]
//     tbl[400 + c*4 + t*2 + s] = log_t  [c][t][s]
//     tbl[800 + c*2 + s]       = log_init[c][s]
// Branch-free WMMA loop: OOB rows/cols are address-CLAMPED, not predicated —
// garbage only lands in discarded D rows (>=100) / cols (>=5).
// ---------------------------------------------------------------------------
__global__ __launch_bounds__(32) void kc_params_kernel(
    const float* __restrict__ kc_reps, const float* __restrict__ W1,
    const float* __restrict__ b1, const float* __restrict__ W2,
    const float* __restrict__ b2, float* __restrict__ tbl)
{
    __shared__ float hsm[16 * 16];
    const int lane  = threadIdx.x;              // 0..31
    const int mloc  = lane & 15;                // tile row / N column
    const int khalf = (lane >> 4) * 2;          // K sub-offset: 0 or 2
    const int m     = blockIdx.x * 16 + mloc;   // chain index c
    const int mrow  = (m < C_N) ? m : (C_N - 1);     // clamp (rows >=100 discarded)
    const int nB    = (mloc < 5) ? mloc : 4;         // clamp (cols >=5 discarded)
    const float* arow = kc_reps + (size_t)mrow * D_N + khalf;
    const float* bcol = W1 + nB;

    v8f acc = {};
#pragma unroll 4
    for (int k0 = 0; k0 < D_N; k0 += 4) {
        const int ka = k0 + khalf;
        v2f a, bm;
        a.x  = arow[k0];
        a.y  = arow[k0 + 1];
        bm.x = bcol[ka * 5];
        bm.y = bcol[(ka + 1) * 5];
        acc = __builtin_amdgcn_wmma_f32_16x16x4_f32(
                  false, a, false, bm, (short)0, acc, false, false);
    }

    // D layout: VGPR j -> (M = j + 8*(lane>>4), N = lane&15)
    const int mhi = (lane >> 4) * 8;
#pragma unroll
    for (int j = 0; j < 8; ++j)
        hsm[(j + mhi) * 16 + mloc] = acc[j];
    __syncthreads();

    if (lane < 16 && m < C_N) {
        const int c = m;
        float h[5];
#pragma unroll
        for (int n = 0; n < 5; ++n)
            h[n] = tanhf(hsm[lane * 16 + n] + b1[n]);
        float kl[5];
#pragma unroll
        for (int q = 0; q < 5; ++q) {
            float s = b2[q];
#pragma unroll
            for (int n = 0; n < 5; ++n) s = fmaf(h[n], W2[n * 5 + q], s);
            kl[q] = s;
        }
        const float l0 = kl[0], l1 = kl[1], l2 = kl[2], l3 = kl[3], l4 = kl[4];

        // obs_logits = [[-l2, l2],[l3, -l3]]; log_softmax over o
        const float M0 = lse2(-l2, l2), M1 = lse2(l3, -l3);
        tbl[c * 4 + 0] = -l2 - M0;
        tbl[c * 4 + 1] =  l2 - M0;
        tbl[c * 4 + 2] =  l3 - M1;
        tbl[c * 4 + 3] = -l3 - M1;
        // trans_logits = [[-l0, l1],[l0, -l1]] ([tgt][src]); softmax over tgt
        const float L0 = lse2(-l0, l0), L1 = lse2(l1, -l1);
        tbl[400 + c * 4 + 0] = -l0 - L0;
        tbl[400 + c * 4 + 1] =  l1 - L1;
        tbl[400 + c * 4 + 2] =  l0 - L0;
        tbl[400 + c * 4 + 3] = -l1 - L1;
        // init_logits = [-l4, l4]
        const float L4 = lse2(-l4, l4);
        tbl[800 + c * 2 + 0] = -l4 - L4;
        tbl[800 + c * 2 + 1] =  l4 - L4;
    }
}

// ---------------------------------------------------------------------------
// Kernel A: E[256000 x 8] = actual_kc[256000 x 100] @ P[100 x 8]
// where P cols 0..3 = log_obs[c][s*2+o], cols 4..7 = log_t[c][t*2+s].
// Pure f32 WMMA GEMM: 16000 M-tiles x 25 k-steps of 16x16x4. Removes the 8
// alpha-independent reductions from the sequential scan's critical path.
// One wave per M-tile, 8 waves per block.
// ---------------------------------------------------------------------------
__global__ __launch_bounds__(256) void blend_gemm_kernel(
    const float* __restrict__ actual_kc, const float* __restrict__ tbl,
    float* __restrict__ E)
{
    const int lane  = threadIdx.x & 31;
    const int wave  = threadIdx.x >> 5;
    const int tile  = blockIdx.x * 8 + wave;    // 0..15999
    const int mloc  = lane & 15;
    const int khalf = (lane >> 4) * 2;
    const size_t r  = (size_t)tile * 16 + mloc; // row in [0, 256000)
    const float* arow = actual_kc + r * C_N + khalf;
    const int nE = mloc & 7;                    // cols 8..15 replicate 0..7 (discarded)
    const float* bcol = (nE < 4) ? (tbl + nE) : (tbl + 400 + (nE - 4));

    v8f acc = {};
#pragma unroll 5
    for (int k0 = 0; k0 < C_N; k0 += 4) {
        const int ka = k0 + khalf;
        v2f a, bm;
        a.x  = arow[k0];
        a.y  = arow[k0 + 1];
        bm.x = bcol[ka * 4];
        bm.y = bcol[(ka + 1) * 4];
        acc = __builtin_amdgcn_wmma_f32_16x16x4_f32(
                  false, a, false, bm, (short)0, acc, false, false);
    }
    const int mhi = (lane >> 4) * 8;
    if (mloc < 8) {
        const size_t rbase = (size_t)tile * 16 + mhi;
#pragma unroll
        for (int j = 0; j < 8; ++j)
            E[(rbase + j) * 8 + mloc] = acc[j];
    }
}

// ---------------------------------------------------------------------------
// Kernel B1 (preferred): HMM forward scan using precomputed E.
// One wave32 per student; lane owns chains c = lane+32j. Only the 2 alpha-
// dependent reductions remain on the serial path. Branch-free tail handling:
// j=0..2 always valid (c<=95); j=3 uses clamped address + 0/1 mask on ch.
// ---------------------------------------------------------------------------
__global__ __launch_bounds__(32) void bkt_scan_lite(
    const int* __restrict__ corr, const float* __restrict__ actual_kc,
    const float* __restrict__ tbl, const float* __restrict__ E,
    float* __restrict__ out)
{
    const int b    = blockIdx.x;
    const int lane = threadIdx.x;
    const int c3   = (lane < 4) ? (96 + lane) : (C_N - 1);   // clamped
    const float m3 = (lane < 4) ? 1.0f : 0.0f;

    float al[4][2];
#pragma unroll
    for (int j = 0; j < 3; ++j) {
        al[j][0] = tbl[800 + (lane + 32 * j) * 2 + 0];
        al[j][1] = tbl[800 + (lane + 32 * j) * 2 + 1];
    }
    al[3][0] = tbl[800 + c3 * 2 + 0];
    al[3][1] = tbl[800 + c3 * 2 + 1];

    const float* ak = actual_kc + (size_t)b * T_N * C_N;
    const float* Eb = E + (size_t)b * T_N * 8;
    const int*   cr = corr + b * T_N;
    float*       ob = out + (size_t)b * T_N * 2;

    for (int t = 0; t < T_N; ++t) {
        if (t + 8 < T_N) {                       // gfx1250 global_prefetch_b8
            __builtin_prefetch(ak + (t + 8) * C_N + lane * 3, 0, 1);
            __builtin_prefetch(Eb + (t + 8) * 8, 0, 1);
        }
        float ch[4];
#pragma unroll
        for (int j = 0; j < 3; ++j)
            ch[j] = ak[t * C_N + lane + 32 * j];
        ch[3] = m3 * ak[t * C_N + c3];

        const float4 e0 = *reinterpret_cast<const float4*>(Eb + t * 8);      // a1[s*2+o]
        const float4 e1 = *reinterpret_cast<const float4*>(Eb + t * 8 + 4);  // tsel[tt*2+s]

        float p0 = 0.0f, p1 = 0.0f;
#pragma unroll
        for (int j = 0; j < 4; ++j) {
            p0 = fmaf(ch[j], al[j][0], p0);
            p1 = fmaf(ch[j], al[j][1], p1);
        }
        const float a20 = wred32(p0), a21 = wred32(p1);

        const float lp0 = lse2(e0.x + a20, e0.z + a21);
        const float lp1 = lse2(e0.y + a20, e0.w + a21);
        const float Z   = lse2(lp0, lp1);
        if (lane == 0)
            *reinterpret_cast<float2*>(ob + 2 * t) = make_float2(lp0 - Z, lp1 - Z);

        const int   y   = cr[t];
        const float oy0 = y ? e0.y : e0.x;
        const float oy1 = y ? e0.w : e0.z;
        const float a30 = lse2(oy0 + a20 + e1.x, oy1 + a21 + e1.y);
        const float a31 = lse2(oy0 + a20 + e1.z, oy1 + a21 + e1.w);
#pragma unroll
        for (int j = 0; j < 4; ++j) {
            al[j][0] = fmaf(ch[j], a30 - al[j][0], al[j][0]);
            al[j][1] = fmaf(ch[j], a31 - al[j][1], al[j][1]);
        }
    }
}

// ---------------------------------------------------------------------------
// Kernel B2 (fallback when d_ws can't hold E): self-contained scan with all
// 10 reductions per step. Branch-free j=3 handling as above.
// ---------------------------------------------------------------------------
__global__ __launch_bounds__(32) void bkt_scan_full(
    const int* __restrict__ corr, const float* __restrict__ actual_kc,
    const float* __restrict__ tbl, float* __restrict__ out)
{
    const int b    = blockIdx.x;
    const int lane = threadIdx.x;
    const int c3   = (lane < 4) ? (96 + lane) : (C_N - 1);
    const float m3 = (lane < 4) ? 1.0f : 0.0f;

    float lo[4][2][2], lt[4][2][2], al[4][2];
    int cidx[4] = { lane, lane + 32, lane + 64, c3 };
#pragma unroll
    for (int j = 0; j < 4; ++j) {
        const int c = cidx[j];
#pragma unroll
        for (int s = 0; s < 2; ++s) {
#pragma unroll
            for (int o = 0; o < 2; ++o) {
                lo[j][s][o] = tbl[      c * 4 + s * 2 + o];
                lt[j][s][o] = tbl[400 + c * 4 + s * 2 + o];
            }
            al[j][s] = tbl[800 + c * 2 + s];
        }
    }

    const float* ak = actual_kc + (size_t)b * T_N * C_N;
    const int*   cr = corr + b * T_N;
    float*       ob = out + (size_t)b * T_N * 2;

    for (int t = 0; t < T_N; ++t) {
        if (t + 8 < T_N)
            __builtin_prefetch(ak + (t + 8) * C_N + lane * 3, 0, 1);

        float ch[4];
#pragma unroll
        for (int j = 0; j < 3; ++j)
            ch[j] = ak[t * C_N + cidx[j]];
        ch[3] = m3 * ak[t * C_N + c3];

        float pa1[2][2] = {{0, 0}, {0, 0}};
        float pt [2][2] = {{0, 0}, {0, 0}};
        float pa2[2]    = {0, 0};
#pragma unroll
        for (int j = 0; j < 4; ++j) {
            const float cj = ch[j];
#pragma unroll
            for (int s = 0; s < 2; ++s) {
                pa2[s] = fmaf(cj, al[j][s], pa2[s]);
#pragma unroll
                for (int o = 0; o < 2; ++o) {
                    pa1[s][o] = fmaf(cj, lo[j][s][o], pa1[s][o]);
                    pt [s][o] = fmaf(cj, lt[j][s][o], pt [s][o]);
                }
            }
        }
        const float a100 = wred32(pa1[0][0]), a101 = wred32(pa1[0][1]);
        const float a110 = wred32(pa1[1][0]), a111 = wred32(pa1[1][1]);
        const float a20  = wred32(pa2[0]),    a21  = wred32(pa2[1]);
        const float t00  = wred32(pt[0][0]),  t01  = wred32(pt[0][1]);
        const float t10  = wred32(pt[1][0]),  t11  = wred32(pt[1][1]);

        const float lp0 = lse2(a100 + a20, a110 + a21);
        const float lp1 = lse2(a101 + a20, a111 + a21);
        const float Z   = lse2(lp0, lp1);
        if (lane == 0)
            *reinterpret_cast<float2*>(ob + 2 * t) = make_float2(lp0 - Z, lp1 - Z);

        const int   y   = cr[t];
        const float oy0 = y ? a101 : a100;
        const float oy1 = y ? a111 : a110;
        const float a30 = lse2(oy0 + a20 + t00, oy1 + a21 + t01);
        const float a31 = lse2(oy0 + a20 + t10, oy1 + a21 + t11);
#pragma unroll
        for (int j = 0; j < 4; ++j) {
            al[j][0] = fmaf(ch[j], a30 - al[j][0], al[j][0]);
            al[j][1] = fmaf(ch[j], a31 - al[j][1], al[j][1]);
        }
    }
}

extern "C" void kernel_launch(void* const* d_in, const int* in_sizes, int n_in,
                              void* d_out, int out_size, void* d_ws, size_t ws_size,
                              hipStream_t stream) {
    const int*   corr      = (const int*)  d_in[0];
    const float* actual_kc = (const float*)d_in[1];
    const float* kc_reps   = (const float*)d_in[2];
    const float* W1        = (const float*)d_in[3];
    const float* b1        = (const float*)d_in[4];
    const float* W2        = (const float*)d_in[5];
    const float* b2        = (const float*)d_in[6];
    // d_in[7] = n_batch_trials (==256, compile-time constant here)

    float* tbl = (float*)d_ws;                       // 1000 floats of tables
    kc_params_kernel<<<(C_N + 15) / 16, 32, 0, stream>>>(kc_reps, W1, b1, W2, b2, tbl);

    const size_t E_elems = (size_t)B_N * T_N * 8;    // 2,048,000 floats (8.2 MB)
    const size_t need    = (1024 + E_elems) * sizeof(float);
    if (ws_size >= need) {
        float* E = tbl + 1024;                       // 4 KB offset, 16B-aligned
        blend_gemm_kernel<<<(B_N * T_N / 16) / 8, 256, 0, stream>>>(actual_kc, tbl, E);
        bkt_scan_lite<<<B_N, 32, 0, stream>>>(corr, actual_kc, tbl, E, (float*)d_out);
    } else {
        bkt_scan_full<<<B_N, 32, 0, stream>>>(corr, actual_kc, tbl, (float*)d_out);
    }
}